// _FallbackGATConv_73186242723982
// MI455X (gfx1250) — compile-verified
//
#include <hip/hip_runtime.h>

#define D_IN   128
#define D_OUT  256           // HEADS * OUT_CH = 4 * 64
#define LDS_STRIDE 130       // 128 + 2 pad: row stride of 2 banks -> conflict-free A reads

typedef __attribute__((ext_vector_type(2))) float v2f;
typedef __attribute__((ext_vector_type(8))) float v8f;

// ---------------------------------------------------------------------------
// Phase 1: edge scatter. One wave per edge iteration: lane l handles the
// float4 at columns [4l, 4l+3]. x gather is a coalesced 512B burst per edge;
// agg/cnt accumulation uses hardware global_atomic_add_f32 (L2-resident).
// ---------------------------------------------------------------------------
__global__ void gat_scatter(const float* __restrict__ x,
                            const long long* __restrict__ src,
                            const long long* __restrict__ dst,
                            float* __restrict__ agg,
                            float* __restrict__ cnt,
                            int nEdges, int wavesTotal) {
  int wave = (int)((blockIdx.x * blockDim.x + threadIdx.x) >> 5);
  int lane = (int)(threadIdx.x & 31);
  for (int e = wave; e < nEdges; e += wavesTotal) {
    int s = (int)src[e];
    int d = (int)dst[e];
    const float4* xr = (const float4*)(x + (size_t)s * D_IN);
    float4 v = xr[lane];
    float* ar = agg + (size_t)d * D_IN + lane * 4;
    unsafeAtomicAdd(ar + 0, v.x);
    unsafeAtomicAdd(ar + 1, v.y);
    unsafeAtomicAdd(ar + 2, v.z);
    unsafeAtomicAdd(ar + 3, v.w);
    if (lane == 0) unsafeAtomicAdd(cnt + d, 1.0f);
  }
}

// ---------------------------------------------------------------------------
// Phase 2: fused normalize + add + GEMM + bias, using V_WMMA_F32_16X16X4_F32.
// Block = 256 threads (8 wave32). Block owns one 16-row stripe of h; wave w
// owns column tiles 2w and 2w+1 (16 cols each -> 256 cols total).
//
// Fragment layouts (ISA 7.12.2, 32-bit):
//   A 16x4 : lanes 0-15 -> {K=0 (v0), K=1 (v1)}; lanes 16-31 -> {K=2, K=3}
//   B 4x16 : same K split; N = lane & 15
//   C/D    : VGPR j -> row j (lanes 0-15) / row j+8 (lanes 16-31), N = lane&15
// ---------------------------------------------------------------------------
__global__ void gat_gemm(const float* __restrict__ x,
                         const float* __restrict__ agg,
                         const float* __restrict__ cnt,
                         const float* __restrict__ W,
                         const float* __restrict__ bias,
                         float* __restrict__ out) {
  __shared__ float hs[16 * LDS_STRIDE];

  const int rt = (int)blockIdx.x;       // row tile: rows [16*rt, 16*rt+15]
  const int t  = (int)threadIdx.x;

  // Cooperative fill of h = x + agg * 1/(cnt + 1e-8): 512 float4s / 256 thr.
#pragma unroll
  for (int i = 0; i < 2; ++i) {
    int lin  = t + i * 256;             // float4 index, 0..511
    int row  = lin >> 5;                // 0..15
    int c4   = lin & 31;                // 0..31
    int grow = rt * 16 + row;
    float inv = 1.0f / (cnt[grow] + 1e-8f);
    float4 xv = ((const float4*)(x   + (size_t)grow * D_IN))[c4];
    float4 av = ((const float4*)(agg + (size_t)grow * D_IN))[c4];
    float* dp = hs + row * LDS_STRIDE + c4 * 4;
    dp[0] = xv.x + av.x * inv;
    dp[1] = xv.y + av.y * inv;
    dp[2] = xv.z + av.z * inv;
    dp[3] = xv.w + av.w * inv;
  }
  __syncthreads();

  const int wv      = t >> 5;           // wave 0..7
  const int lane    = t & 31;
  const int halfsel = lane >> 4;        // 0: K+{0,1}, 1: K+{2,3}
  const int lrow    = lane & 15;        // A row / B col / D col
  const int ct0     = wv * 2;
  const int ct1     = wv * 2 + 1;

  const float* hrow = hs + lrow * LDS_STRIDE;
  v8f acc0 = {};
  v8f acc1 = {};

#pragma unroll 8
  for (int k0 = 0; k0 < D_IN; k0 += 4) {
    int kk = k0 + halfsel * 2;
    v2f a;
    a.x = hrow[kk];
    a.y = hrow[kk + 1];
    v2f b0, b1;
    b0.x = W[(size_t)kk       * D_OUT + ct0 * 16 + lrow];
    b0.y = W[(size_t)(kk + 1) * D_OUT + ct0 * 16 + lrow];
    b1.x = W[(size_t)kk       * D_OUT + ct1 * 16 + lrow];
    b1.y = W[(size_t)(kk + 1) * D_OUT + ct1 * 16 + lrow];
    acc0 = __builtin_amdgcn_wmma_f32_16x16x4_f32(false, a, false, b0,
                                                 (short)0, acc0, false, false);
    acc1 = __builtin_amdgcn_wmma_f32_16x16x4_f32(false, a, false, b1,
                                                 (short)0, acc1, false, false);
  }

  float bs0 = bias[ct0 * 16 + lrow];
  float bs1 = bias[ct1 * 16 + lrow];
  int rbase = rt * 16 + halfsel * 8;
#pragma unroll
  for (int j = 0; j < 8; ++j) {
    size_t ro = (size_t)(rbase + j) * D_OUT;
    out[ro + ct0 * 16 + lrow] = acc0[j] + bs0;
    out[ro + ct1 * 16 + lrow] = acc1[j] + bs1;
  }
}

// ---------------------------------------------------------------------------
// Inputs: d_in[0]=x [N*128 f32], d_in[1]=edge_index [2*E i64],
//         d_in[2]=W [128*256 f32], d_in[3]=b [256 f32].
// d_ws: agg [N*128 f32] then cnt [N f32].
// ---------------------------------------------------------------------------
extern "C" void kernel_launch(void* const* d_in, const int* in_sizes, int n_in,
                              void* d_out, int out_size, void* d_ws, size_t ws_size,
                              hipStream_t stream) {
  const float*     x  = (const float*)d_in[0];
  const long long* ei = (const long long*)d_in[1];
  const float*     W  = (const float*)d_in[2];
  const float*     b  = (const float*)d_in[3];
  float*           out = (float*)d_out;

  const int nEdges = in_sizes[1] / 2;
  const int nNodes = in_sizes[0] / D_IN;       // 100000, multiple of 16

  float* agg = (float*)d_ws;
  float* cnt = agg + (size_t)nNodes * D_IN;

  hipMemsetAsync(d_ws, 0, ((size_t)nNodes * D_IN + nNodes) * sizeof(float), stream);

  const long long* src = ei;
  const long long* dst = ei + nEdges;

  const int scatterBlocks = 4096;              // 32768 waves, ~98 edges each
  gat_scatter<<<scatterBlocks, 256, 0, stream>>>(x, src, dst, agg, cnt,
                                                 nEdges, scatterBlocks * 8);

  gat_gemm<<<nNodes / 16, 256, 0, stream>>>(x, agg, cnt, W, b, out);
}